// MultiheadSIM_29910152249948
// MI455X (gfx1250) — compile-verified
//
#include <hip/hip_runtime.h>
#include <cmath>

typedef float v2f __attribute__((ext_vector_type(2)));
typedef float v8f __attribute__((ext_vector_type(8)));

#define D_MODEL 256
#define N_PATCH 128
#define N_HEADS 8
#define D_K     32
#define BATCH   32

__device__ __forceinline__ float gelu_tanh(float x) {
    float x3 = x * x * x;
    float t = tanhf(0.7978845608028654f * (x + 0.044715f * x3));
    return 0.5f * x * (1.0f + t);
}

// ---------------------------------------------------------------------------
// Generic fp32 WMMA GEMM: OUT[M,256] = act(A[M,256] @ W[256,256] + bias)
// One 16x16 output tile per wave, K stepped by 4 via v_wmma_f32_16x16x4_f32.
// act: 0 = none, 1 = tanh-GELU
// ---------------------------------------------------------------------------
__global__ void wmma_gemm_256(const float* __restrict__ A,
                              const float* __restrict__ W,
                              const float* __restrict__ bias,
                              float* __restrict__ OUT,
                              int M, int act) {
    const int lane = threadIdx.x & 31;
    const int wave = threadIdx.x >> 5;
    const int half = lane >> 4;    // 0: K=0,1   1: K=2,3
    const int r    = lane & 15;

    const int tilesN = D_MODEL / 16;              // 16
    const int nTiles = (M / 16) * tilesN;
    int tile = blockIdx.x * (blockDim.x >> 5) + wave;
    if (tile >= nTiles) return;                    // wave-uniform

    const int row0 = (tile / tilesN) * 16;
    const int col0 = (tile % tilesN) * 16;

    v8f acc = {};
    const float* arow = A + (size_t)(row0 + r) * D_MODEL;
    #pragma unroll 4
    for (int k = 0; k < D_MODEL; k += 4) {
        const int ka = k + 2 * half;
        v2f a, b;
        a.x = arow[ka];
        a.y = arow[ka + 1];
        b.x = W[(size_t)ka * D_MODEL + col0 + r];
        b.y = W[(size_t)(ka + 1) * D_MODEL + col0 + r];
        acc = __builtin_amdgcn_wmma_f32_16x16x4_f32(false, a, false, b,
                                                    (short)0, acc, false, false);
    }

    const float bval = bias ? bias[col0 + r] : 0.0f;
    #pragma unroll
    for (int g = 0; g < 8; ++g) {
        const int row = row0 + g + 8 * half;
        float v = acc[g] + bval;
        if (act == 1) v = gelu_tanh(v);
        OUT[(size_t)row * D_MODEL + col0 + r] = v;
    }
}

// ---------------------------------------------------------------------------
// feats[(h*128+n)*256 + d] = memory_w[n*2048 + h*256 + d]
// ---------------------------------------------------------------------------
__global__ void pack_feats(const float* __restrict__ mw, float* __restrict__ feats) {
    int idx = blockIdx.x * blockDim.x + threadIdx.x;
    if (idx >= N_HEADS * N_PATCH * D_MODEL) return;
    int d = idx & 255;
    int n = (idx >> 8) & 127;
    int h = idx >> 15;
    feats[idx] = mw[(size_t)n * (N_HEADS * D_MODEL) + h * D_MODEL + d];
}

// ---------------------------------------------------------------------------
// conn[h,i,j] = 1 iff (logits1 - logits0) + g1 - g0 > 0, where
// logits_diff = relu(As[h,j,:] + Ar[h,i,:]) . (w_cat[:,1]-w_cat[:,0]) + db
// (fc_out_b is folded into As; argmax is invariant to temp/softmax)
// Block = (h*128 + i), 128 threads, thread j.
// ---------------------------------------------------------------------------
__global__ void conn_kernel(const float* __restrict__ As,
                            const float* __restrict__ Ar,
                            const float* __restrict__ fc_cat_w,
                            const float* __restrict__ fc_cat_b,
                            const float* __restrict__ gumbel_u,
                            float* __restrict__ conn) {
    __shared__ float s_ar[D_MODEL];
    __shared__ float s_wd[D_MODEL];
    const int hi = blockIdx.x;          // h*128 + i
    const int h  = hi >> 7;
    const int i  = hi & 127;
    const int j  = threadIdx.x;

    for (int c = threadIdx.x; c < D_MODEL; c += 128) {
        s_ar[c] = Ar[(size_t)hi * D_MODEL + c];
        s_wd[c] = fc_cat_w[c * 2 + 1] - fc_cat_w[c * 2 + 0];
    }
    __syncthreads();

    const float* as = As + ((size_t)h * N_PATCH + j) * D_MODEL;
    float acc = 0.0f;
    for (int c = 0; c < D_MODEL; ++c) {
        float hv = as[c] + s_ar[c];
        hv = hv > 0.0f ? hv : 0.0f;
        acc = fmaf(hv, s_wd[c], acc);
    }
    const float ldiff = acc + (fc_cat_b[1] - fc_cat_b[0]);

    const size_t gidx = (((size_t)h * N_PATCH + i) * N_PATCH + j) * 2;
    const float u0 = gumbel_u[gidx];
    const float u1 = gumbel_u[gidx + 1];
    const float g0 = -logf(-logf(u0 + 1e-10f) + 1e-10f);
    const float g1 = -logf(-logf(u1 + 1e-10f) + 1e-10f);

    conn[((size_t)h * N_PATCH + i) * N_PATCH + j] =
        (ldiff + g1 > g0) ? 1.0f : 0.0f;
}

// ---------------------------------------------------------------------------
// Attention: block = (b, h, row-quarter rb). 256 threads = 8 waves.
//   scores(32x128) = scale * q kT + (conn-1)*1e9  -> softmax -> @ v (32x32)
// WMMA for both matmuls; scores staged in padded LDS for the row softmax.
// ---------------------------------------------------------------------------
__global__ void attn_kernel(const float* __restrict__ Q,
                            const float* __restrict__ K,
                            const float* __restrict__ V,
                            const float* __restrict__ conn,
                            float* __restrict__ OUT) {
    __shared__ float s_q[32][D_K];              //  4 KB
    __shared__ float s_k[N_PATCH][D_K];         // 16 KB
    __shared__ float s_v[N_PATCH][D_K];         // 16 KB
    __shared__ float s_s[32][N_PATCH + 4];      // ~17 KB, padded vs bank conflicts

    const int bh = blockIdx.x >> 2;
    const int rb = blockIdx.x & 3;              // row quarter
    const int b  = bh >> 3;
    const int h  = bh & 7;
    const int tid  = threadIdx.x;
    const int lane = tid & 31;
    const int wave = tid >> 5;
    const int half = lane >> 4;
    const int r    = lane & 15;
    const int base = b * N_PATCH;

    for (int idx = tid; idx < N_PATCH * D_K; idx += 256) {
        const int n  = idx >> 5;
        const int dk = idx & 31;
        const size_t g = ((size_t)(base + n)) * D_MODEL + h * D_K + dk;
        s_k[n][dk] = K[g];
        s_v[n][dk] = V[g];
        if (n < 32) {
            const size_t gq = ((size_t)(base + rb * 32 + n)) * D_MODEL + h * D_K + dk;
            s_q[n][dk] = Q[gq];
        }
    }
    __syncthreads();

    const float scale = 0.17677669529663687f;   // 32^-0.5

    // ---- scores: M=32, N=128 -> 2x8 = 16 tiles, 2 per wave ----
    for (int t = 0; t < 2; ++t) {
        const int tileId = wave * 2 + t;
        const int ti = tileId >> 3, tj = tileId & 7;
        const int row0 = ti * 16, col0 = tj * 16;
        v8f acc = {};
        #pragma unroll
        for (int kk = 0; kk < D_K; kk += 4) {
            const int ka = kk + 2 * half;
            v2f a, bb;
            a.x  = s_q[row0 + r][ka];  a.y  = s_q[row0 + r][ka + 1];
            bb.x = s_k[col0 + r][ka];  bb.y = s_k[col0 + r][ka + 1];
            acc = __builtin_amdgcn_wmma_f32_16x16x4_f32(false, a, false, bb,
                                                        (short)0, acc, false, false);
        }
        #pragma unroll
        for (int g = 0; g < 8; ++g) {
            const int il = row0 + g + 8 * half;      // local row (0..31)
            const int j  = col0 + r;
            const int ig = rb * 32 + il;             // global row
            const float m = conn[((size_t)h * N_PATCH + ig) * N_PATCH + j];
            s_s[il][j] = acc[g] * scale + (m - 1.0f) * 1e9f;
        }
    }
    __syncthreads();

    // ---- row softmax (32 rows; wave 0) ----
    if (tid < 32) {
        const int row = tid;
        float mx = -3.0e38f;
        for (int j = 0; j < N_PATCH; ++j) mx = fmaxf(mx, s_s[row][j]);
        float sum = 0.0f;
        for (int j = 0; j < N_PATCH; ++j) {
            const float e = expf(s_s[row][j] - mx);
            s_s[row][j] = e;
            sum += e;
        }
        const float inv = 1.0f / sum;
        for (int j = 0; j < N_PATCH; ++j) s_s[row][j] *= inv;
    }
    __syncthreads();

    // ---- attn @ v: M=32, N=32 -> 4 tiles, waves 0..3 (wave-uniform branch) ----
    if (wave < 4) {
        const int row0 = (wave >> 1) * 16;
        const int col0 = (wave & 1) * 16;
        v8f acc = {};
        for (int kk = 0; kk < N_PATCH; kk += 4) {
            const int ka = kk + 2 * half;
            v2f a, bb;
            a.x  = s_s[row0 + r][ka];   a.y  = s_s[row0 + r][ka + 1];
            bb.x = s_v[ka][col0 + r];   bb.y = s_v[ka + 1][col0 + r];
            acc = __builtin_amdgcn_wmma_f32_16x16x4_f32(false, a, false, bb,
                                                        (short)0, acc, false, false);
        }
        #pragma unroll
        for (int g = 0; g < 8; ++g) {
            const int ig = rb * 32 + row0 + g + 8 * half;
            OUT[((size_t)(base + ig)) * D_MODEL + h * D_K + col0 + r] = acc[g];
        }
    }
}

// ---------------------------------------------------------------------------
extern "C" void kernel_launch(void* const* d_in, const int* in_sizes, int n_in,
                              void* d_out, int out_size, void* d_ws, size_t ws_size,
                              hipStream_t stream) {
    (void)in_sizes; (void)n_in; (void)out_size; (void)ws_size;

    const float* x        = (const float*)d_in[0];
    const float* gumbel_u = (const float*)d_in[1];
    const float* memory_w = (const float*)d_in[2];
    const float* fc_out_w = (const float*)d_in[3];
    const float* fc_out_b = (const float*)d_in[4];
    const float* fc_cat_w = (const float*)d_in[5];
    const float* fc_cat_b = (const float*)d_in[6];
    const float* wq = (const float*)d_in[7];
    const float* bq = (const float*)d_in[8];
    const float* wk = (const float*)d_in[9];
    const float* bk = (const float*)d_in[10];
    const float* wv = (const float*)d_in[11];
    const float* bv = (const float*)d_in[12];
    const float* out_w = (const float*)d_in[13];
    const float* out_b = (const float*)d_in[14];
    const float* mlp_w1 = (const float*)d_in[15];
    const float* mlp_b1 = (const float*)d_in[16];
    const float* mlp_w2 = (const float*)d_in[17];
    const float* mlp_b2 = (const float*)d_in[18];
    float* outp = (float*)d_out;

    // workspace layout (floats)
    float* ws      = (float*)d_ws;
    float* feats   = ws;                      // 1024*256
    float* As      = feats   + 262144;        // 1024*256
    float* Ar      = As      + 262144;        // 1024*256
    float* conn    = Ar      + 262144;        // 8*128*128
    float* tmp1    = conn    + 131072;        // 4096*256
    float* xm      = tmp1    + 1048576;
    float* q       = xm      + 1048576;
    float* k       = q       + 1048576;
    float* v       = k       + 1048576;
    float* attnout = v       + 1048576;

    const int M  = BATCH * N_PATCH;           // 4096
    const int Mf = N_HEADS * N_PATCH;         // 1024

    // --- connection-mask branch (batch independent) ---
    pack_feats<<<(Mf * D_MODEL + 255) / 256, 256, 0, stream>>>(memory_w, feats);
    wmma_gemm_256<<<(Mf / 16) * 16 / 8, 256, 0, stream>>>(
        feats, fc_out_w,                     fc_out_b, As, Mf, 0);
    wmma_gemm_256<<<(Mf / 16) * 16 / 8, 256, 0, stream>>>(
        feats, fc_out_w + D_MODEL * D_MODEL, nullptr,  Ar, Mf, 0);
    conn_kernel<<<N_HEADS * N_PATCH, 128, 0, stream>>>(
        As, Ar, fc_cat_w, fc_cat_b, gumbel_u, conn);

    // --- patch-integration MLP ---
    wmma_gemm_256<<<(M / 16) * 16 / 8, 256, 0, stream>>>(x,    mlp_w1, mlp_b1, tmp1, M, 1);
    wmma_gemm_256<<<(M / 16) * 16 / 8, 256, 0, stream>>>(tmp1, mlp_w2, mlp_b2, xm,   M, 0);

    // --- q, k, v projections ---
    wmma_gemm_256<<<(M / 16) * 16 / 8, 256, 0, stream>>>(xm, wq, bq, q, M, 0);
    wmma_gemm_256<<<(M / 16) * 16 / 8, 256, 0, stream>>>(xm, wk, bk, k, M, 0);
    wmma_gemm_256<<<(M / 16) * 16 / 8, 256, 0, stream>>>(xm, wv, bv, v, M, 0);

    // --- masked multi-head attention ---
    attn_kernel<<<BATCH * N_HEADS * 4, 256, 0, stream>>>(q, k, v, conn, attnout);

    // --- output projection ---
    wmma_gemm_256<<<(M / 16) * 16 / 8, 256, 0, stream>>>(attnout, out_w, out_b, outp, M, 0);
}